// DiffPoolEncoder_8761733283914
// MI455X (gfx1250) — compile-verified
//
#include <hip/hip_runtime.h>
#include <hip/hip_bf16.h>
#include <math.h>

// ---- problem constants (match reference) ----
static constexpr int Bg   = 32;    // graphs
static constexpr int NPG  = 128;   // nodes per graph
static constexpr int Nn   = Bg * NPG;   // 4096 total nodes
static constexpr int Dd   = 64;    // feature dim
static constexpr int HID  = 64;
static constexpr int HEADS= 4;
static constexpr int DH   = HID / HEADS;  // 16
static constexpr int P0   = 32;
static constexpr int P1   = 8;
static constexpr int DLIN = 256;
static constexpr int DOUT = 128;
static constexpr int Ee   = 65536;
static constexpr int MLPH = 64;

typedef __attribute__((ext_vector_type(2))) float v2f;
typedef __attribute__((ext_vector_type(8))) float v8f;

// -------------------------------------------------------------------------
// Utility kernels
// -------------------------------------------------------------------------
__global__ void k_zero(float* __restrict__ p, int n) {
    int i = blockIdx.x * blockDim.x + threadIdx.x;
    if (i < n) p[i] = 0.0f;
}

// A[src,dst] = 1 within each graph's 128x128 block
__global__ void k_build_adj(const int* __restrict__ ei, float* __restrict__ Ablk, int E) {
    int i = blockIdx.x * blockDim.x + threadIdx.x;
    if (i < E) {
        int s = ei[i];
        int d = ei[E + i];
        int g = s / NPG;
        Ablk[(size_t)g * NPG * NPG + (size_t)(s % NPG) * NPG + (d % NPG)] = 1.0f;
    }
}

template <int P>
__global__ void k_row_softmax(float* __restrict__ X, int M) {
    int r = blockIdx.x * blockDim.x + threadIdx.x;
    if (r >= M) return;
    float* row = X + (size_t)r * P;
    float mx = -3.0e38f;
    #pragma unroll 8
    for (int p = 0; p < P; ++p) mx = fmaxf(mx, row[p]);
    float s = 0.0f;
    #pragma unroll 8
    for (int p = 0; p < P; ++p) { float e = expf(row[p] - mx); row[p] = e; s += e; }
    float inv = 1.0f / s;
    #pragma unroll 8
    for (int p = 0; p < P; ++p) row[p] *= inv;
}

__global__ void k_meanpool(const float* __restrict__ X2, float* __restrict__ g,
                           int Bn, int P, int D) {
    int i = blockIdx.x * blockDim.x + threadIdx.x;
    if (i < Bn * D) {
        int b = i / D, d = i % D;
        float s = 0.0f;
        for (int p = 0; p < P; ++p) s += X2[((size_t)b * P + p) * D + d];
        g[i] = s / (float)P;
    }
}

// -------------------------------------------------------------------------
// Batched GEMM via V_WMMA_F32_16X16X4_F32 (exact fp32), fully templated so
// all strides are immediate offsets (A pair merges to b64 loads, B second
// element folds into the load offset; no 64-bit muls/adds in the hot loop).
// One wave computes a 16x32 C tile (two v8f accumulators, shared A fragment).
// Out-of-range rows/cols use CLAMPED loads (valid finite data) that only
// pollute C elements the guarded stores never write -> K-loop is branch-free
// with EXEC all-1s.  Requires K % 16 == 0 (true for every GEMM here).
// -------------------------------------------------------------------------
template <int M, int N, int K, int LDA, int LDB, int LDC,
          bool TRANSA, bool RELU, bool HASBIAS>
__global__ __launch_bounds__(32)
void k_gemm_wmma(const float* __restrict__ A, const float* __restrict__ Bm,
                 const float* __restrict__ bias, float* __restrict__ C,
                 int sA, int sB, int sC) {
    int lane = threadIdx.x;
    int col0 = blockIdx.x * 32, row0 = blockIdx.y * 16;
    const float* Ab = A  + (size_t)blockIdx.z * sA;
    const float* Bb = Bm + (size_t)blockIdx.z * sB;
    float*       Cb = C  + (size_t)blockIdx.z * sC;

    int half = lane >> 4;      // 0: K pair {k,k+1}; 1: {k+2,k+3}
    int h2   = half * 2;
    int l    = lane & 15;      // M index (A frag) / N index (B frag) in tile
    int m    = row0 + l;
    int cn0  = col0 + l;
    int cn1  = cn0 + 16;
    bool nok0 = cn0 < N, nok1 = cn1 < N;
    int mc  = (m < M) ? m : (M - 1);     // clamped: loads always valid/finite
    int cc0 = nok0 ? cn0 : (N - 1);
    int cc1 = nok1 ? cn1 : (N - 1);

    const float* pA  = Ab + (TRANSA ? (h2 * LDA + mc) : (mc * LDA + h2));
    const float* pB0 = Bb + h2 * LDB + cc0;
    const float* pB1 = Bb + h2 * LDB + cc1;
    constexpr int DA1 = TRANSA ? LDA : 1;        // a0 -> a1 element step
    constexpr int DAK = TRANSA ? 4 * LDA : 4;    // per K-step advance
    constexpr int DBK = 4 * LDB;

    v8f acc0 = {0.f, 0.f, 0.f, 0.f, 0.f, 0.f, 0.f, 0.f};
    v8f acc1 = {0.f, 0.f, 0.f, 0.f, 0.f, 0.f, 0.f, 0.f};
    #pragma unroll 4
    for (int ks = 0; ks < K / 4; ++ks) {
        v2f av  = {pA[0],  pA[DA1]};
        v2f bv0 = {pB0[0], pB0[LDB]};
        v2f bv1 = {pB1[0], pB1[LDB]};
        acc0 = __builtin_amdgcn_wmma_f32_16x16x4_f32(
            false, av, false, bv0, (short)0, acc0, false, false);
        acc1 = __builtin_amdgcn_wmma_f32_16x16x4_f32(
            false, av, false, bv1, (short)0, acc1, false, false);
        pA += DAK; pB0 += DBK; pB1 += DBK;
    }

    float bs0 = HASBIAS ? bias[cc0] : 0.0f;
    float bs1 = HASBIAS ? bias[cc1] : 0.0f;
    #pragma unroll
    for (int vg = 0; vg < 8; ++vg) {
        int rr = row0 + vg + half * 8;   // D layout: VGPR r -> M = r (+8 upper half)
        if (rr < M) {
            if (nok0) {
                float v = acc0[vg] + bs0;
                if (RELU) v = fmaxf(v, 0.0f);
                Cb[(size_t)rr * LDC + cn0] = v;
            }
            if (nok1) {
                float v = acc1[vg] + bs1;
                if (RELU) v = fmaxf(v, 0.0f);
                Cb[(size_t)rr * LDC + cn1] = v;
            }
        }
    }
}

template <int M, int N, int K, int LDA, int LDB, int LDC,
          bool TRANSA, bool RELU, bool HASBIAS>
static inline void gemm(const float* A, const float* Bm, const float* bias, float* C,
                        int batch, int sA, int sB, int sC, hipStream_t stream) {
    k_gemm_wmma<M, N, K, LDA, LDB, LDC, TRANSA, RELU, HASBIAS>
        <<<dim3((N + 31) / 32, (M + 15) / 16, batch), dim3(32), 0, stream>>>(
            A, Bm, bias, C, sA, sB, sC);
}

// -------------------------------------------------------------------------
// GAT attention: one block per (head, graph); blockDim.x == NV (128 or 32).
// Hp: [batch*NV, HEADS*DH].  adj: [batch, NV, NV] (0/1-ish floats).
// Hout[(g*NV+i), h*DH+d] = elu( softmax_j(mask? lrelu(es_i+ed_j) : -1e9) @ Hp )
// -------------------------------------------------------------------------
template <int NV>
__global__ __launch_bounds__(NV)
void k_gat_attn(const float* __restrict__ Hp, const float* __restrict__ adj,
                const float* __restrict__ a_src, const float* __restrict__ a_dst,
                float* __restrict__ Hout) {
    __shared__ float HpL[NV * DH];
    __shared__ float esL[NV];
    __shared__ float edL[NV];

    int h = blockIdx.x, g = blockIdx.y, i = threadIdx.x;
    const float* hrow = Hp + ((size_t)(g * NV + i)) * (HEADS * DH) + h * DH;
    float es = 0.0f, ed = 0.0f;
    #pragma unroll
    for (int d = 0; d < DH; ++d) {
        float v = hrow[d];
        HpL[i * DH + d] = v;
        es += v * a_src[h * DH + d];
        ed += v * a_dst[h * DH + d];
    }
    esL[i] = es; edL[i] = ed;
    __syncthreads();

    const float* arow = adj + ((size_t)g * NV + i) * NV;
    float esi = esL[i];

    float mx = -3.0e38f;
    #pragma unroll 4
    for (int j = 0; j < NV; ++j) {
        float x = esi + edL[j];
        float e = (x >= 0.0f) ? x : 0.2f * x;               // leaky_relu 0.2
        bool  msk = (arow[j] > 0.0f) || (j == i);           // adjacency + self loop
        float val = msk ? e : -1.0e9f;
        mx = fmaxf(mx, val);
    }
    float acc[DH];
    #pragma unroll
    for (int d = 0; d < DH; ++d) acc[d] = 0.0f;
    float sum = 0.0f;
    #pragma unroll 2
    for (int j = 0; j < NV; ++j) {
        float x = esi + edL[j];
        float e = (x >= 0.0f) ? x : 0.2f * x;
        bool  msk = (arow[j] > 0.0f) || (j == i);
        float val = msk ? e : -1.0e9f;
        float p = expf(val - mx);                            // masked -> exactly 0
        sum += p;
        const float* hj = &HpL[j * DH];
        #pragma unroll
        for (int d = 0; d < DH; ++d) acc[d] += p * hj[d];
    }
    float inv = 1.0f / sum;
    float* orow = Hout + ((size_t)(g * NV + i)) * (HEADS * DH) + h * DH;
    #pragma unroll
    for (int d = 0; d < DH; ++d) {
        float v = acc[d] * inv;
        orow[d] = (v > 0.0f) ? v : expm1f(v);                // elu(alpha=1)
    }
}

// -------------------------------------------------------------------------
// Host-side helpers
// -------------------------------------------------------------------------
struct EncP {
    const float *W0, *W1, *as0, *as1, *ad0, *ad1, *mW0, *mW1, *mb0, *mb1;
};

template <int BATCH, int NV, int POUT>
static void run_encoder(const EncP& e, const float* X, const float* adj,
                        float* bufHp, float* bufH1, float* bufH2,
                        float* out, hipStream_t stream) {
    constexpr int M = BATCH * NV;
    // layer 1: Hp = X @ W0  [M,64]x[64,64]
    gemm<M, HID, Dd, Dd, HID, HID, false, false, false>(
        X, e.W0, nullptr, bufHp, 1, 0, 0, 0, stream);
    k_gat_attn<NV><<<dim3(HEADS, BATCH), dim3(NV), 0, stream>>>(bufHp, adj, e.as0, e.ad0, bufH1);
    // layer 2: Hp = H1 @ W1
    gemm<M, HID, HID, HID, HID, HID, false, false, false>(
        bufH1, e.W1, nullptr, bufHp, 1, 0, 0, 0, stream);
    k_gat_attn<NV><<<dim3(HEADS, BATCH), dim3(NV), 0, stream>>>(bufHp, adj, e.as1, e.ad1, bufH2);
    // MLP: relu(H2 @ mW0 + mb0) @ mW1 + mb1
    gemm<M, MLPH, HID, HID, MLPH, MLPH, false, true, true>(
        bufH2, e.mW0, e.mb0, bufHp, 1, 0, 0, 0, stream);
    gemm<M, POUT, MLPH, MLPH, POUT, POUT, false, false, true>(
        bufHp, e.mW1, e.mb1, out, 1, 0, 0, 0, stream);
}

extern "C" void kernel_launch(void* const* d_in, const int* in_sizes, int n_in,
                              void* d_out, int out_size, void* d_ws, size_t ws_size,
                              hipStream_t stream) {
    const float* x  = (const float*)d_in[0];
    const int*   ei = (const int*)d_in[1];
    auto enc = [&](int base) {
        EncP p;
        p.W0  = (const float*)d_in[base + 0];  p.W1  = (const float*)d_in[base + 1];
        p.as0 = (const float*)d_in[base + 2];  p.as1 = (const float*)d_in[base + 3];
        p.ad0 = (const float*)d_in[base + 4];  p.ad1 = (const float*)d_in[base + 5];
        p.mW0 = (const float*)d_in[base + 6];  p.mW1 = (const float*)d_in[base + 7];
        p.mb0 = (const float*)d_in[base + 8];  p.mb1 = (const float*)d_in[base + 9];
        return p;
    };
    EncP pool0 = enc(2), emb0 = enc(12), pool1 = enc(22), emb1 = enc(32);
    const float* linW1 = (const float*)d_in[42];
    const float* linb1 = (const float*)d_in[43];
    const float* linW2 = (const float*)d_in[44];
    const float* linb2 = (const float*)d_in[45];

    float* ws = (float*)d_ws;
    size_t off = 0;
    auto alloc = [&](size_t nf) { float* p = ws + off; off += nf; return p; };
    float* Ablk  = alloc((size_t)Bg * NPG * NPG);   // 0/1 adjacency blocks
    float* bufHp = alloc((size_t)Nn * HID);
    float* bufH1 = alloc((size_t)Nn * HID);
    float* bufH2 = alloc((size_t)Nn * HID);
    float* S     = alloc((size_t)Nn * P0);
    float* Z     = alloc((size_t)Nn * Dd);
    float* T     = alloc((size_t)Bg * NPG * P0);
    float* A1    = alloc((size_t)Bg * P0 * P0);
    float* X1    = alloc((size_t)Bg * P0 * Dd);
    float* S1    = alloc((size_t)Bg * P0 * P1);
    float* Z1    = alloc((size_t)Bg * P0 * Dd);
    float* X2    = alloc((size_t)Bg * P1 * Dd);
    float* gbuf  = alloc((size_t)Bg * Dd);
    float* hbuf  = alloc((size_t)Bg * DLIN);

    // ---- adjacency ----
    int nA = Bg * NPG * NPG;
    k_zero<<<(nA + 255) / 256, 256, 0, stream>>>(Ablk, nA);
    k_build_adj<<<(Ee + 255) / 256, 256, 0, stream>>>(ei, Ablk, Ee);

    // ---- level 0 ----
    run_encoder<Bg, NPG, P0>(pool0, x, Ablk, bufHp, bufH1, bufH2, S, stream);
    k_row_softmax<P0><<<(Nn + 255) / 256, 256, 0, stream>>>(S, Nn);
    run_encoder<Bg, NPG, Dd>(emb0, x, Ablk, bufHp, bufH1, bufH2, Z, stream);

    // X1[g] = S[g]^T @ Z[g]   (M=32, N=64, K=128)
    gemm<P0, Dd, NPG, P0, Dd, Dd, true, false, false>(
        S, Z, nullptr, X1, Bg, NPG * P0, NPG * Dd, P0 * Dd, stream);
    // T[g] = Ablk[g] @ S[g]   (M=128, N=32, K=128)
    gemm<NPG, P0, NPG, NPG, P0, P0, false, false, false>(
        Ablk, S, nullptr, T, Bg, NPG * NPG, NPG * P0, NPG * P0, stream);
    // A1[g] = S[g]^T @ T[g]   (M=32, N=32, K=128)
    gemm<P0, P0, NPG, P0, P0, P0, true, false, false>(
        S, T, nullptr, A1, Bg, NPG * P0, NPG * P0, P0 * P0, stream);

    // ---- level 1 ----
    run_encoder<Bg, P0, P1>(pool1, X1, A1, bufHp, bufH1, bufH2, S1, stream);
    k_row_softmax<P1><<<(Bg * P0 + 255) / 256, 256, 0, stream>>>(S1, Bg * P0);
    run_encoder<Bg, P0, Dd>(emb1, X1, A1, bufHp, bufH1, bufH2, Z1, stream);

    // X2[g] = S1[g]^T @ Z1[g]  (M=8, N=64, K=32)
    gemm<P1, Dd, P0, P1, Dd, Dd, true, false, false>(
        S1, Z1, nullptr, X2, Bg, P0 * P1, P0 * Dd, P1 * Dd, stream);

    // global mean pool over P1
    k_meanpool<<<(Bg * Dd + 255) / 256, 256, 0, stream>>>(X2, gbuf, Bg, P1, Dd);

    // final MLP: relu(g @ W1 + b1) @ W2 + b2 -> d_out [32,128]
    gemm<Bg, DLIN, Dd, Dd, DLIN, DLIN, false, true, true>(
        gbuf, linW1, linb1, hbuf, 1, 0, 0, 0, stream);
    gemm<Bg, DOUT, DLIN, DLIN, DOUT, DOUT, false, false, true>(
        hbuf, linW2, linb2, (float*)d_out, 1, 0, 0, 0, stream);
}